// RNAFoldModel_34187939676352
// MI455X (gfx1250) — compile-verified
//
#include <hip/hip_runtime.h>
#include <hip/hip_bf16.h>
#include <math.h>

typedef __attribute__((ext_vector_type(2))) float v2f;
typedef __attribute__((ext_vector_type(8))) float v8f;

#define LSEQ 1024

// ---------------- wave reductions (wave32) ----------------
__device__ inline float wsum(float v) {
    for (int o = 16; o > 0; o >>= 1) v += __shfl_xor(v, o, 32);
    return v;
}
__device__ inline float wmaxr(float v) {
    for (int o = 16; o > 0; o >>= 1) v = fmaxf(v, __shfl_xor(v, o, 32));
    return v;
}

// ---------------- Stage 1: A1 = [seq@W_seq + b_seq + PE | mean_n(msa@W_msa)+b_msa]
__global__ void build_A1_kernel(const float* __restrict__ seq, const float* __restrict__ msa,
                                const float* __restrict__ Wseq, const float* __restrict__ bseq,
                                const float* __restrict__ Wmsa, const float* __restrict__ bmsa,
                                float* __restrict__ A1) {
    int l = blockIdx.x * 256 + threadIdx.x;
    if (l >= LSEQ) return;
    // one-hot counts over MSA depth
    float cnt0 = 0.f, cnt1 = 0.f, cnt2 = 0.f, cnt3 = 0.f, cnt4 = 0.f;
    for (int n = 0; n < 128; ++n) {
        const float* p = msa + (size_t)n * (LSEQ * 5) + l * 5;
        cnt0 += p[0]; cnt1 += p[1]; cnt2 += p[2]; cnt3 += p[3]; cnt4 += p[4];
    }
    float sv[5];
    const float* q = seq + l * 5;
    sv[0] = q[0]; sv[1] = q[1]; sv[2] = q[2]; sv[3] = q[3]; sv[4] = q[4];

    const float c = logf(10000.f) * (1.f / 32.f);  // 2/64 * ln(1e4)
    float* Arow = A1 + (size_t)l * 96;
    for (int d = 0; d < 64; ++d) {
        float s = bseq[d];
        for (int v = 0; v < 5; ++v) s += sv[v] * Wseq[v * 64 + d];
        int k = d & 31;
        float invk = expf(c * (float)k);
        float arg = (float)l / invk;
        s += (d < 32) ? sinf(arg) : cosf(arg);
        Arow[d] = s;
    }
    for (int e = 0; e < 32; ++e) {
        float s = cnt0 * Wmsa[0 * 32 + e] + cnt1 * Wmsa[1 * 32 + e] + cnt2 * Wmsa[2 * 32 + e]
                + cnt3 * Wmsa[3 * 32 + e] + cnt4 * Wmsa[4 * 32 + e];
        Arow[64 + e] = s * (1.f / 128.f) + bmsa[e];
    }
}

// ---------------- WMMA f32 GEMM: C(1024x64) = A(1024x96) @ B(96x64) + bias [+ addsrc]
// one wave per 16-row tile, 4 column tiles of 16; 128 threads/block -> 4 waves
__global__ void gemm96x64_kernel(const float* __restrict__ A, const float* __restrict__ B,
                                 const float* __restrict__ bias, const float* __restrict__ addsrc,
                                 float* __restrict__ C) {
    int wave = threadIdx.x >> 5;
    int lane = threadIdx.x & 31;
    int row0 = (blockIdx.x * 4 + wave) * 16;
    int m = lane & 15;
    int koff = (lane >> 4) << 1;     // 0 or 2

    v8f acc0 = {}, acc1 = {}, acc2 = {}, acc3 = {};
    for (int kb = 0; kb < 96; kb += 4) {
        v2f a;
        const float* Ar = A + (size_t)(row0 + m) * 96 + kb + koff;
        a.x = Ar[0];
        a.y = Ar[1];
        const float* Br0 = B + (size_t)(kb + koff) * 64;
        const float* Br1 = Br0 + 64;
        v2f b0, b1, b2, b3;
        b0.x = Br0[m];        b0.y = Br1[m];
        b1.x = Br0[16 + m];   b1.y = Br1[16 + m];
        b2.x = Br0[32 + m];   b2.y = Br1[32 + m];
        b3.x = Br0[48 + m];   b3.y = Br1[48 + m];
        acc0 = __builtin_amdgcn_wmma_f32_16x16x4_f32(false, a, false, b0, (short)0, acc0, false, false);
        acc1 = __builtin_amdgcn_wmma_f32_16x16x4_f32(false, a, false, b1, (short)0, acc1, false, false);
        acc2 = __builtin_amdgcn_wmma_f32_16x16x4_f32(false, a, false, b2, (short)0, acc2, false, false);
        acc3 = __builtin_amdgcn_wmma_f32_16x16x4_f32(false, a, false, b3, (short)0, acc3, false, false);
    }
    int rb = (lane >> 4) * 8;
#pragma unroll
    for (int rr = 0; rr < 8; ++rr) {
        int row = row0 + rb + rr;
        float* Crow = C + (size_t)row * 64;
        float a0 = bias[m],      a1 = bias[16 + m], a2 = bias[32 + m], a3 = bias[48 + m];
        if (addsrc) {
            const float* S = addsrc + (size_t)row * 64;
            a0 += S[m]; a1 += S[16 + m]; a2 += S[32 + m]; a3 += S[48 + m];
        }
        Crow[m]      = acc0[rr] + a0;
        Crow[16 + m] = acc1[rr] + a1;
        Crow[32 + m] = acc2[rr] + a2;
        Crow[48 + m] = acc3[rr] + a3;
    }
}

// ---------------- WMMA pi/pj: pi(1024x32)=node1@W_pair[0:64], pj=node1@W_pair[64:128]
__global__ void gemm_pair_kernel(const float* __restrict__ node1, const float* __restrict__ Wpair,
                                 float* __restrict__ pi, float* __restrict__ pj) {
    int wave = threadIdx.x >> 5;
    int lane = threadIdx.x & 31;
    int row0 = (blockIdx.x * 4 + wave) * 16;
    int m = lane & 15;
    int koff = (lane >> 4) << 1;

    v8f p0 = {}, p1 = {}, q0 = {}, q1 = {};
    for (int kb = 0; kb < 64; kb += 4) {
        v2f a;
        const float* Ar = node1 + (size_t)(row0 + m) * 64 + kb + koff;
        a.x = Ar[0];
        a.y = Ar[1];
        const float* B1r0 = Wpair + (size_t)(kb + koff) * 32;           // Wi rows
        const float* B1r1 = B1r0 + 32;
        const float* B2r0 = Wpair + (size_t)(64 + kb + koff) * 32;      // Wj rows
        const float* B2r1 = B2r0 + 32;
        v2f b0, b1, b2, b3;
        b0.x = B1r0[m];       b0.y = B1r1[m];
        b1.x = B1r0[16 + m];  b1.y = B1r1[16 + m];
        b2.x = B2r0[m];       b2.y = B2r1[m];
        b3.x = B2r0[16 + m];  b3.y = B2r1[16 + m];
        p0 = __builtin_amdgcn_wmma_f32_16x16x4_f32(false, a, false, b0, (short)0, p0, false, false);
        p1 = __builtin_amdgcn_wmma_f32_16x16x4_f32(false, a, false, b1, (short)0, p1, false, false);
        q0 = __builtin_amdgcn_wmma_f32_16x16x4_f32(false, a, false, b2, (short)0, q0, false, false);
        q1 = __builtin_amdgcn_wmma_f32_16x16x4_f32(false, a, false, b3, (short)0, q1, false, false);
    }
    int rb = (lane >> 4) * 8;
#pragma unroll
    for (int rr = 0; rr < 8; ++rr) {
        int row = row0 + rb + rr;
        pi[(size_t)row * 32 + m]      = p0[rr];
        pi[(size_t)row * 32 + 16 + m] = p1[rr];
        pj[(size_t)row * 32 + m]      = q0[rr];
        pj[(size_t)row * 32 + 16 + m] = q1[rr];
    }
}

// ---------------- mean over rows of pj (1024x32) ----------------
__global__ void meanpj_kernel(const float* __restrict__ pj, float* __restrict__ meanpj) {
    __shared__ float sh[256];
    int t = threadIdx.x;
    int col = t & 31, g = t >> 5;
    float s = 0.f;
    for (int r = g * 128; r < g * 128 + 128; ++r) s += pj[(size_t)r * 32 + col];
    sh[t] = s;
    __syncthreads();
    if (t < 32) {
        float tot = 0.f;
        for (int k = 0; k < 8; ++k) tot += sh[k * 32 + t];
        meanpj[t] = tot * (1.f / 1024.f);
    }
}

// ---------------- A2 = [node1 | pi + mean_pj + b_pair] ----------------
__global__ void build_A2_kernel(const float* __restrict__ node1, const float* __restrict__ pi,
                                const float* __restrict__ meanpj, const float* __restrict__ bpair,
                                float* __restrict__ A2) {
    int idx = blockIdx.x * 256 + threadIdx.x;   // 1024*96 elements
    int l = idx / 96, c = idx - l * 96;
    float v;
    if (c < 64) v = node1[(size_t)l * 64 + c];
    else {
        int e = c - 64;
        v = pi[(size_t)l * 32 + e] + meanpj[e] + bpair[e];
    }
    A2[idx] = v;
}

// ---------------- LN + mag + tj, one wave per row ----------------
__global__ void finalize_kernel(const float* __restrict__ node2, const float* __restrict__ lns,
                                const float* __restrict__ lno, const float* __restrict__ Wmag,
                                const float* __restrict__ bmag, const float* __restrict__ pj,
                                const float* __restrict__ Wattn,
                                float* __restrict__ mag, float* __restrict__ tj) {
    int lane = threadIdx.x & 31;
    int row = blockIdx.x * 8 + (threadIdx.x >> 5);
    const float* r = node2 + (size_t)row * 64;
    float x0 = r[lane], x1 = r[lane + 32];
    float mu = wsum(x0 + x1) * (1.f / 64.f);
    float d0 = x0 - mu, d1 = x1 - mu;
    float var = wsum(d0 * d0 + d1 * d1) * (1.f / 64.f);
    float rs = rsqrtf(var + 1e-5f);
    float y0 = d0 * rs * lns[lane] + lno[lane];
    float y1 = d1 * rs * lns[lane + 32] + lno[lane + 32];
    float dot = wsum(y0 * Wmag[lane] + y1 * Wmag[lane + 32]);
    float tjv = wsum(pj[(size_t)row * 32 + lane] * Wattn[lane]);
    if (lane == 0) {
        mag[row] = tanhf(dot + bmag[0]) * 0.1f;
        tj[row] = tjv;
    }
}

// ---------------- analytic helix coords, centered ----------------
__global__ void coords_kernel(float* __restrict__ coords) {
    __shared__ float sx[1024], sy[1024], sz[1024];
    int i = threadIdx.x;
    const float rot = (float)(32.7 * 3.14159265358979323846 / 180.0);
    float th = (float)i * rot;
    float x = 10.f * cosf(th), y = 10.f * sinf(th), z = 2.8f * (float)i;
    sx[i] = x; sy[i] = y; sz[i] = z;
    __syncthreads();
    for (int s = 512; s > 0; s >>= 1) {
        if (i < s) { sx[i] += sx[i + s]; sy[i] += sy[i + s]; sz[i] += sz[i + s]; }
        __syncthreads();
    }
    float mx = sx[0] * (1.f / 1024.f), my = sy[0] * (1.f / 1024.f), mz = sz[0] * (1.f / 1024.f);
    coords[3 * i + 0] = x - mx;
    coords[3 * i + 1] = y - my;
    coords[3 * i + 2] = z - mz;
}

// ---------------- row-softmax over j with directions, one block per i ----------------
__global__ void attn_kernel(const float* __restrict__ tj, const float* __restrict__ coords,
                            const float* __restrict__ mag, const float* __restrict__ Wattn,
                            float* __restrict__ out) {
    __shared__ float stj[1024], sx[1024], sy[1024], sz[1024];
    __shared__ float red[32];
    __shared__ float Msh;
    int i = blockIdx.x, t = threadIdx.x;
    float wd = Wattn[32];
    for (int j = t; j < 1024; j += 256) {
        stj[j] = tj[j];
        sx[j] = coords[3 * j + 0];
        sy[j] = coords[3 * j + 1];
        sz[j] = coords[3 * j + 2];
    }
    __syncthreads();
    float cix = sx[i], ciy = sy[i], ciz = sz[i];

    // pass 1: row max
    float mx = -1e30f;
    for (int j = t; j < 1024; j += 256) {
        float dx = cix - sx[j], dy = ciy - sy[j], dz = ciz - sz[j];
        float dist = sqrtf(dx * dx + dy * dy + dz * dz) + 1e-8f;
        mx = fmaxf(mx, stj[j] + wd * dist);
    }
    mx = wmaxr(mx);
    if ((t & 31) == 0) red[t >> 5] = mx;
    __syncthreads();
    if (t == 0) {
        float mm = red[0];
        for (int w = 1; w < 8; ++w) mm = fmaxf(mm, red[w]);
        Msh = mm;
    }
    __syncthreads();
    float M = Msh;

    // pass 2: softmax-weighted direction sum
    float s = 0.f, vx = 0.f, vy = 0.f, vz = 0.f;
    for (int j = t; j < 1024; j += 256) {
        float dx = cix - sx[j], dy = ciy - sy[j], dz = ciz - sz[j];
        float dist = sqrtf(dx * dx + dy * dy + dz * dz) + 1e-8f;
        float w = expf(stj[j] + wd * dist - M);
        s += w;
        float inv = w / dist;
        vx += inv * dx; vy += inv * dy; vz += inv * dz;
    }
    s = wsum(s); vx = wsum(vx); vy = wsum(vy); vz = wsum(vz);
    if ((t & 31) == 0) {
        int w = t >> 5;
        red[w * 4 + 0] = s; red[w * 4 + 1] = vx; red[w * 4 + 2] = vy; red[w * 4 + 3] = vz;
    }
    __syncthreads();
    if (t == 0) {
        float S = 0.f, X = 0.f, Y = 0.f, Z = 0.f;
        for (int w = 0; w < 8; ++w) {
            S += red[w * 4 + 0]; X += red[w * 4 + 1]; Y += red[w * 4 + 2]; Z += red[w * 4 + 3];
        }
        float mg = mag[i];
        float ux = mg * X / S, uy = mg * Y / S, uz = mg * Z / S;
        out[3 * i + 0] = fminf(fmaxf(cix + ux, -999.999f), 9999.999f);
        out[3 * i + 1] = fminf(fmaxf(ciy + uy, -999.999f), 9999.999f);
        out[3 * i + 2] = fminf(fmaxf(ciz + uz, -999.999f), 9999.999f);
    }
}

extern "C" void kernel_launch(void* const* d_in, const int* in_sizes, int n_in,
                              void* d_out, int out_size, void* d_ws, size_t ws_size,
                              hipStream_t stream) {
    const float* seq   = (const float*)d_in[0];   // (1024,5)
    const float* msa   = (const float*)d_in[1];   // (128,1024,5)
    const float* Wseq  = (const float*)d_in[2];   // (5,64)
    const float* bseq  = (const float*)d_in[3];   // (64)
    const float* Wmsa  = (const float*)d_in[4];   // (5,32)
    const float* bmsa  = (const float*)d_in[5];   // (32)
    const float* Wmp   = (const float*)d_in[6];   // (96,64)
    const float* bmp   = (const float*)d_in[7];   // (64)
    const float* Wpair = (const float*)d_in[8];   // (160,32)
    const float* bpair = (const float*)d_in[9];   // (32)
    const float* Wnode = (const float*)d_in[10];  // (96,64)
    const float* bnode = (const float*)d_in[11];  // (64)
    const float* lns   = (const float*)d_in[12];  // (64)
    const float* lno   = (const float*)d_in[13];  // (64)
    const float* Wattn = (const float*)d_in[14];  // (33,1)
    // d_in[15] = b_attn (cancels in softmax)
    const float* Wmag  = (const float*)d_in[16];  // (64,1)
    const float* bmag  = (const float*)d_in[17];  // (1)
    float* out = (float*)d_out;                   // (1024,3)

    float* w = (float*)d_ws;
    float* A1     = w;                 // 1024*96
    float* node1  = A1 + 1024 * 96;    // 1024*64
    float* pi     = node1 + 1024 * 64; // 1024*32
    float* pj     = pi + 1024 * 32;    // 1024*32
    float* A2     = pj + 1024 * 32;    // 1024*96
    float* node2  = A2 + 1024 * 96;    // 1024*64
    float* magb   = node2 + 1024 * 64; // 1024
    float* tjb    = magb + 1024;       // 1024
    float* meanpj = tjb + 1024;        // 32
    float* coords = meanpj + 32;       // 1024*3

    build_A1_kernel<<<4, 256, 0, stream>>>(seq, msa, Wseq, bseq, Wmsa, bmsa, A1);
    gemm96x64_kernel<<<16, 128, 0, stream>>>(A1, Wmp, bmp, nullptr, node1);
    gemm_pair_kernel<<<16, 128, 0, stream>>>(node1, Wpair, pi, pj);
    meanpj_kernel<<<1, 256, 0, stream>>>(pj, meanpj);
    build_A2_kernel<<<384, 256, 0, stream>>>(node1, pi, meanpj, bpair, A2);
    gemm96x64_kernel<<<16, 128, 0, stream>>>(A2, Wnode, bnode, node1, node2);
    finalize_kernel<<<128, 256, 0, stream>>>(node2, lns, lno, Wmag, bmag, pj, Wattn, magb, tjb);
    coords_kernel<<<1, 1024, 0, stream>>>(coords);
    attn_kernel<<<1024, 256, 0, stream>>>(tjb, coords, magb, Wattn, out);
}